// VectorQuantizer_19825569038732
// MI455X (gfx1250) — compile-verified
//
#include <hip/hip_runtime.h>

typedef __attribute__((ext_vector_type(2))) float v2f;
typedef __attribute__((ext_vector_type(8))) float v8f;

#define DIMS   64
#define CODES  512
#define NROWS  65536
#define NELEM  (NROWS * DIMS)

// ---------------------------------------------------------------------------
// Kernel A: enorm[k] = sum_d E[d][k]^2  -> ws[0..511]; ws[512] = 0 (loss acc)
// ---------------------------------------------------------------------------
__global__ void vq_enorm_init(const float* __restrict__ E, float* __restrict__ ws) {
    const int k = threadIdx.x;           // 512 threads
    float s = 0.0f;
#pragma unroll 8
    for (int d = 0; d < DIMS; ++d) {
        const float e = E[d * CODES + k];
        s += e * e;
    }
    ws[k] = s;
    if (k == 0) ws[CODES] = 0.0f;
}

// ---------------------------------------------------------------------------
// Kernel B: per 16-row tile (one wave): fp32 WMMA scores, argmin, gather,
//           straight-through output, loss partial sum.
// ---------------------------------------------------------------------------
__global__ void __launch_bounds__(256) vq_main(const float* __restrict__ X,
                                               const float* __restrict__ E,
                                               float* __restrict__ out,
                                               float* __restrict__ ws) {
    __shared__ int s_idx[8][16];

    const int tid  = threadIdx.x;
    const int lane = tid & 31;
    const int wave = tid >> 5;
    const int rowBase = (blockIdx.x * 8 + wave) * 16;

    // A-matrix 16x4 f32 layout: lanes 0..15 -> rows 0..15, K = {0,1};
    // lanes 16..31 -> rows 0..15, K = {2,3}.
    const int r    = lane & 15;
    const int koff = (lane >> 4) << 1;

    // Preload the whole 16x64 x-tile as 16 A fragments (v2f each).
    v2f a[16];
    const float* xrow = X + (rowBase + r) * DIMS + koff;
#pragma unroll
    for (int g = 0; g < 16; ++g)
        a[g] = *reinterpret_cast<const v2f*>(xrow + g * 4);

    // B fragments come straight from E (128 KB, hot in cache).
    // B 4x16 f32 layout (mirror of A): lanes 0..15 -> K={0,1}, lanes 16..31 -> K={2,3}.
    const float* ebase = E + koff * CODES + r;

    float minval[8];
    int   minidx[8];
#pragma unroll
    for (int i = 0; i < 8; ++i) { minval[i] = 3.4e38f; minidx[i] = 0; }

    for (int c = 0; c < CODES / 16; ++c) {
        const int   col = c * 16 + r;        // this lane's N column (global code id)
        const float en  = ws[col];           // ||e_col||^2
        const float* ep = ebase + c * 16;

        v8f acc = {};
#pragma unroll
        for (int g = 0; g < 16; ++g) {       // K = 64 in steps of 4
            v2f b;
            b.x = ep[g * 4 * CODES];
            b.y = ep[g * 4 * CODES + CODES];
            acc = __builtin_amdgcn_wmma_f32_16x16x4_f32(
                false, a[g], false, b, (short)0, acc, false, false);
        }

        // C/D layout: VGPR i, lanes 0..15 -> row i, lanes 16..31 -> row 8+i;
        // column for this lane is `col` in every VGPR.
#pragma unroll
        for (int i = 0; i < 8; ++i) {
            const float dist = en - 2.0f * acc[i];   // + ||x||^2 is argmin-invariant
            if (dist < minval[i]) { minval[i] = dist; minidx[i] = col; }
        }
    }

    // Min-with-index reduce across each 16-lane half (rows 0..7 in lanes 0..15,
    // rows 8..15 in lanes 16..31). Tie-break to lowest index (argmin semantics).
#pragma unroll
    for (int mask = 8; mask >= 1; mask >>= 1) {
#pragma unroll
        for (int i = 0; i < 8; ++i) {
            const float ov = __shfl_xor(minval[i], mask, 32);
            const int   oi = __shfl_xor(minidx[i], mask, 32);
            if (ov < minval[i] || (ov == minval[i] && oi < minidx[i])) {
                minval[i] = ov; minidx[i] = oi;
            }
        }
    }
    if (lane == 0) {
#pragma unroll
        for (int i = 0; i < 8; ++i) s_idx[wave][i] = minidx[i];
    }
    if (lane == 16) {
#pragma unroll
        for (int i = 0; i < 8; ++i) s_idx[wave][8 + i] = minidx[i];
    }
    __syncthreads();

    // Gather winning code columns, emit out = x + (q - x) (reference rounding),
    // accumulate (x - q)^2 for the loss.
    float sum = 0.0f;
#pragma unroll 4
    for (int m = 0; m < 16; ++m) {
        const int k   = s_idx[wave][m];
        const int row = rowBase + m;
#pragma unroll
        for (int h = 0; h < 2; ++h) {
            const int   d  = lane + h * 32;
            const float q  = E[d * CODES + k];
            const float xv = X[row * DIMS + d];
            const float t  = q - xv;
            out[row * DIMS + d] = xv + t;     // straight-through value
            const float df = xv - q;
            sum += df * df;
        }
    }
#pragma unroll
    for (int mask = 16; mask >= 1; mask >>= 1)
        sum += __shfl_xor(sum, mask, 32);
    if (lane == 0) atomicAdd(ws + CODES, sum);
}

// ---------------------------------------------------------------------------
// Kernel C: loss = (1 + 0.25) * mean((x - q)^2)
// ---------------------------------------------------------------------------
__global__ void vq_loss(const float* __restrict__ ws, float* __restrict__ out) {
    out[NELEM] = 1.25f * ws[CODES] / (float)NELEM;
}

extern "C" void kernel_launch(void* const* d_in, const int* in_sizes, int n_in,
                              void* d_out, int out_size, void* d_ws, size_t ws_size,
                              hipStream_t stream) {
    const float* X = (const float*)d_in[0];   // (64,32,32,64) f32
    const float* E = (const float*)d_in[1];   // (64,512) f32
    float* out = (float*)d_out;               // 4194304 outputs + 1 loss
    float* ws  = (float*)d_ws;                // [0..511] enorm, [512] loss acc

    vq_enorm_init<<<1, CODES, 0, stream>>>(E, ws);
    vq_main<<<NROWS / (16 * 8), 256, 0, stream>>>(X, E, out, ws);
    vq_loss<<<1, 1, 0, stream>>>(ws, out);
}